// AttentionHead_71064528880180
// MI455X (gfx1250) — compile-verified
//
#include <hip/hip_runtime.h>

// CDNA5 / gfx1250, wave32. fp32 attention via V_WMMA_F32_16X16X4_F32,
// memory->LDS staging via GLOBAL_LOAD_ASYNC_TO_LDS_B128 (ASYNCcnt).
typedef __attribute__((ext_vector_type(2))) float v2f;
typedef __attribute__((ext_vector_type(8))) float v8f;
typedef __attribute__((ext_vector_type(4))) int   v4i;

typedef __attribute__((address_space(1))) v4i* gptr_v4i;  // global
typedef __attribute__((address_space(3))) v4i* lptr_v4i;  // LDS

#define SEQ   4096
#define DIM   1024
#define HEAD  64
#define NB    4

#if __has_builtin(__builtin_amdgcn_global_load_async_to_lds_b128) && \
    __has_builtin(__builtin_amdgcn_s_wait_asynccnt)
#define HAVE_ASYNC_LDS 1
#endif

#ifdef HAVE_ASYNC_LDS
#define ASYNC_WAIT(n) __builtin_amdgcn_s_wait_asynccnt(n)
#else
#define ASYNC_WAIT(n) ((void)0)
#endif

// Copy 16 bytes global -> LDS. Async path: no VGPR round trip, tracked on
// ASYNCcnt; completions are in issue order (CDNA5 ISA).
__device__ __forceinline__ void copy16_to_lds(const float* g, float* l) {
#ifdef HAVE_ASYNC_LDS
    __builtin_amdgcn_global_load_async_to_lds_b128(
        (gptr_v4i)(g), (lptr_v4i)(l), 0, 0);
#else
    *reinterpret_cast<float4*>(l) = *reinterpret_cast<const float4*>(g);
#endif
}

// ---------------------------------------------------------------------------
// Projection: out[m,n] = sum_d x[m,d] * W[d,n];  M=16384, N=64, K=1024.
// grid = (M/16, 3), block = 128 (4 waves). Wave w computes n-tile w (16 cols).
// x staged through LDS in 16x256 chunks, row stride 260 floats
// (260 % 64 == 4 -> conflict-free; 1040B rows keep 16B alignment).
// ---------------------------------------------------------------------------
__global__ __launch_bounds__(128) void proj_kernel(
    const float* __restrict__ x,
    const float* __restrict__ Wq, const float* __restrict__ Wk,
    const float* __restrict__ Wv,
    float* __restrict__ qb, float* __restrict__ kb, float* __restrict__ vb)
{
    const float* W; float* out;
    if (blockIdx.y == 0)      { W = Wq; out = qb; }
    else if (blockIdx.y == 1) { W = Wk; out = kb; }
    else                      { W = Wv; out = vb; }

    __shared__ float xs[16 * 260];

    const int mt   = blockIdx.x;
    const int wave = threadIdx.x >> 5;
    const int lane = threadIdx.x & 31;
    const int hf   = lane >> 4;      // which 16-lane half
    const int ll   = lane & 15;
    const int n0   = wave * 16;

    v8f acc = {};
    for (int chunk = 0; chunk < 4; ++chunk) {
        __syncthreads();
        // stage x[mt*16 .. +16][chunk*256 .. +256]: 8 async b128 per thread
        for (int i = threadIdx.x; i < 16 * 64; i += 128) {
            int r = i >> 6, c4 = i & 63;
            copy16_to_lds(x + (size_t)(mt * 16 + r) * DIM + chunk * 256 + c4 * 4,
                          &xs[r * 260 + c4 * 4]);
        }
        ASYNC_WAIT(0);
        __syncthreads();

        const float* Wc = W + (size_t)(chunk * 256) * HEAD;
        for (int d = 0; d < 256; d += 4) {
            // A 16x4: lane = M row, VGPR0/1 = K d+{0,1} (hi half: d+{2,3})
            v2f a, b;
            a.x = xs[ll * 260 + d + 2 * hf];
            a.y = xs[ll * 260 + d + 2 * hf + 1];
            // B 4x16: lane = N col, VGPR0/1 = K d+{0,1} (hi half: d+{2,3})
            b.x = Wc[(size_t)(d + 2 * hf) * HEAD + n0 + ll];
            b.y = Wc[(size_t)(d + 2 * hf + 1) * HEAD + n0 + ll];
            acc = __builtin_amdgcn_wmma_f32_16x16x4_f32(
                false, a, false, b, (short)0, acc, false, false);
        }
    }
    // C layout: VGPR v -> row v (lanes 0-15) / row v+8 (lanes 16-31)
    #pragma unroll
    for (int v = 0; v < 8; ++v) {
        int m = mt * 16 + v + 8 * hf;
        out[(size_t)m * HEAD + n0 + ll] = acc[v];
    }
}

// ---------------------------------------------------------------------------
// Flash attention: grid = (SEQ/64, B), block = 128 (4 waves).
// Block owns 64 query rows; wave owns 16. Key blocks of 64, causal.
// Async staging trick: issue K copies, then V copies; s_wait_asynccnt(8)
// releases K while V streams in behind QK^T + softmax; s_wait_asynccnt(0)
// at the (already required) P-routing barrier releases V for PV.
// LDS row stride 68 (68 % 64 == 4 -> conflict-free, 272B rows 16B-aligned).
// ---------------------------------------------------------------------------
__global__ __launch_bounds__(128) void attn_kernel(
    const float* __restrict__ qb, const float* __restrict__ kb,
    const float* __restrict__ vb, float* __restrict__ out)
{
    __shared__ float Ks[64 * 68];
    __shared__ float Vs[64 * 68];
    __shared__ float Ps[4][16 * 68];

    const int qblk = blockIdx.x;
    const int b    = blockIdx.y;
    const int wave = threadIdx.x >> 5;
    const int lane = threadIdx.x & 31;
    const int hf   = lane >> 4;
    const int ll   = lane & 15;

    const size_t base = (size_t)b * SEQ * HEAD;
    const int qrow0 = qblk * 64 + wave * 16;

    // Q fragments (A layout), kept in registers for the whole key loop
    v2f qf[16];
    {
        const float* qp = qb + base + (size_t)(qrow0 + ll) * HEAD;
        #pragma unroll
        for (int i = 0; i < 16; ++i) {
            qf[i].x = qp[4 * i + 2 * hf];
            qf[i].y = qp[4 * i + 2 * hf + 1];
        }
    }

    v8f o[4] = {};                 // 16x64 output accumulator
    float mrow[8], lrow[8];
    #pragma unroll
    for (int v = 0; v < 8; ++v) { mrow[v] = -3.0e38f; lrow[v] = 0.0f; }

    for (int j = 0; j <= qblk; ++j) {
        __syncthreads();   // all waves done reading Ks/Vs/Ps of iter j-1
        {
            const float* kp = kb + base + (size_t)j * 64 * HEAD;
            const float* vp = vb + base + (size_t)j * 64 * HEAD;
            // 8 K-copies per thread first...
            for (int i = threadIdx.x; i < 64 * 16; i += 128) {
                int r = i >> 4, c4 = i & 15;
                copy16_to_lds(kp + (size_t)r * HEAD + c4 * 4, &Ks[r * 68 + c4 * 4]);
            }
            // ...then 8 V-copies (complete later, in order)
            for (int i = threadIdx.x; i < 64 * 16; i += 128) {
                int r = i >> 4, c4 = i & 15;
                copy16_to_lds(vp + (size_t)r * HEAD + c4 * 4, &Vs[r * 68 + c4 * 4]);
            }
        }
        ASYNC_WAIT(8);     // first 8 (K) complete; V still in flight
        __syncthreads();   // K visible to all waves

        // S = Q @ K^T : 4 tiles of 16x16, contracted over HEAD=64
        v8f sf[4];
        #pragma unroll
        for (int kc = 0; kc < 4; ++kc) {
            v8f s = {};
            #pragma unroll
            for (int d = 0; d < 64; d += 4) {
                v2f bf;        // B[k=d][n=key] = K[key][d]
                bf.x = Ks[(kc * 16 + ll) * 68 + d + 2 * hf];
                bf.y = Ks[(kc * 16 + ll) * 68 + d + 2 * hf + 1];
                s = __builtin_amdgcn_wmma_f32_16x16x4_f32(
                    false, qf[d >> 2], false, bf, (short)0, s, false, false);
            }
            sf[kc] = s;
        }

        // scale + causal mask
        #pragma unroll
        for (int kc = 0; kc < 4; ++kc) {
            int col = j * 64 + kc * 16 + ll;
            #pragma unroll
            for (int v = 0; v < 8; ++v) {
                int row = qrow0 + v + 8 * hf;
                float sv = sf[kc][v] * 0.125f;   // 1/sqrt(64)
                sf[kc][v] = (col > row) ? -3.0e38f : sv;
            }
        }

        // online softmax: rows live per-VGPR-slot; reduce across 16-lane half
        #pragma unroll
        for (int v = 0; v < 8; ++v) {
            float t = fmaxf(fmaxf(sf[0][v], sf[1][v]), fmaxf(sf[2][v], sf[3][v]));
            t = fmaxf(t, __shfl_xor(t, 1, 32));
            t = fmaxf(t, __shfl_xor(t, 2, 32));
            t = fmaxf(t, __shfl_xor(t, 4, 32));
            t = fmaxf(t, __shfl_xor(t, 8, 32));
            float mnew = fmaxf(mrow[v], t);
            float c = __expf(mrow[v] - mnew);
            float rs = 0.0f;
            #pragma unroll
            for (int kc = 0; kc < 4; ++kc) {
                float p = __expf(sf[kc][v] - mnew);
                sf[kc][v] = p;
                rs += p;
            }
            rs += __shfl_xor(rs, 1, 32);
            rs += __shfl_xor(rs, 2, 32);
            rs += __shfl_xor(rs, 4, 32);
            rs += __shfl_xor(rs, 8, 32);
            lrow[v] = lrow[v] * c + rs;
            mrow[v] = mnew;
            #pragma unroll
            for (int t4 = 0; t4 < 4; ++t4) o[t4][v] *= c;
        }

        // route P through LDS: C layout -> A layout
        #pragma unroll
        for (int kc = 0; kc < 4; ++kc)
            #pragma unroll
            for (int v = 0; v < 8; ++v)
                Ps[wave][(v + 8 * hf) * 68 + kc * 16 + ll] = sf[kc][v];

        ASYNC_WAIT(0);     // V landed in LDS
        __syncthreads();   // V + all P tiles visible

        // O += P @ V  (contract over 64 keys)
        #pragma unroll
        for (int t4 = 0; t4 < 4; ++t4) {
            v8f oacc = o[t4];
            #pragma unroll
            for (int k = 0; k < 64; k += 4) {
                v2f a, bf;
                a.x  = Ps[wave][ll * 68 + k + 2 * hf];
                a.y  = Ps[wave][ll * 68 + k + 2 * hf + 1];
                bf.x = Vs[(k + 2 * hf) * 68 + t4 * 16 + ll];
                bf.y = Vs[(k + 2 * hf + 1) * 68 + t4 * 16 + ll];
                oacc = __builtin_amdgcn_wmma_f32_16x16x4_f32(
                    false, a, false, bf, (short)0, oacc, false, false);
            }
            o[t4] = oacc;
        }
    }

    #pragma unroll
    for (int v = 0; v < 8; ++v) {
        float rinv = 1.0f / lrow[v];
        int row = qrow0 + v + 8 * hf;
        #pragma unroll
        for (int t4 = 0; t4 < 4; ++t4)
            out[base + (size_t)row * HEAD + t4 * 16 + ll] = o[t4][v] * rinv;
    }
}

// ---------------------------------------------------------------------------
extern "C" void kernel_launch(void* const* d_in, const int* in_sizes, int n_in,
                              void* d_out, int out_size, void* d_ws, size_t ws_size,
                              hipStream_t stream) {
    // setup_inputs order: x, Wk, Wq, Wv
    const float* x  = (const float*)d_in[0];
    const float* Wk = (const float*)d_in[1];
    const float* Wq = (const float*)d_in[2];
    const float* Wv = (const float*)d_in[3];
    float* out = (float*)d_out;

    float* qb = (float*)d_ws;                       // 16384 x 64
    float* kb = qb + (size_t)NB * SEQ * HEAD;
    float* vb = kb + (size_t)NB * SEQ * HEAD;       // total 12 MB workspace

    dim3 pgrid(NB * SEQ / 16, 3);                   // 1024 m-tiles x {Q,K,V}
    proj_kernel<<<pgrid, 128, 0, stream>>>(x, Wq, Wk, Wv, qb, kb, vb);

    dim3 agrid(SEQ / 64, NB);                       // 64 q-blocks x 4 batches
    attn_kernel<<<agrid, 128, 0, stream>>>(qb, kb, vb, out);
}